// NodeModel_63402307223697
// MI455X (gfx1250) — compile-verified
//
#include <hip/hip_runtime.h>

typedef __attribute__((ext_vector_type(2))) float v2f;
typedef __attribute__((ext_vector_type(8))) float v8f;

#define N_NODES 50000
#define N_EDGES 800000
#define DIM     64
#define IN_DIM  192

// ---------------- Kernel 0: zero the aggregation buffer ----------------
__global__ __launch_bounds__(256) void zero_kernel(float4* p, int n4) {
  int i = blockIdx.x * blockDim.x + threadIdx.x;
  if (i < n4) {
    float4 z; z.x = 0.f; z.y = 0.f; z.z = 0.f; z.w = 0.f;
    p[i] = z;
  }
}

// ---------------- Kernel 1: scatter-add edge_attr into agg ----------------
// one thread per (edge, 4-float chunk): 800000 * 16 threads
__global__ __launch_bounds__(256) void scatter_kernel(const float* __restrict__ edge_attr,
                                                      const int*   __restrict__ col,
                                                      float*       __restrict__ agg) {
  int i = blockIdx.x * blockDim.x + threadIdx.x;
  if (i >= N_EDGES * 16) return;
  int e  = i >> 4;
  int c4 = (i & 15) << 2;
  int c  = col[e];
  const float4 v = *(const float4*)(edge_attr + (size_t)e * DIM + c4);
  float* dst = agg + (size_t)c * DIM + c4;
  unsafeAtomicAdd(dst + 0, v.x);
  unsafeAtomicAdd(dst + 1, v.y);
  unsafeAtomicAdd(dst + 2, v.z);
  unsafeAtomicAdd(dst + 3, v.w);
}

// ---------------- Kernel 2: transpose weights (W[k][n] -> WT[n][k]) ----------------
__global__ __launch_bounds__(256) void transpose_kernel(const float* __restrict__ W1,
                                                        const float* __restrict__ W2,
                                                        float* __restrict__ W1T,
                                                        float* __restrict__ W2T) {
  int i = blockIdx.x * blockDim.x + threadIdx.x;
  if (i < IN_DIM * DIM) {
    int k = i / DIM, n = i % DIM;
    W1T[(size_t)n * IN_DIM + k] = W1[i];
  }
  if (i < DIM * DIM) {
    int k = i / DIM, n = i % DIM;
    W2T[(size_t)n * DIM + k] = W2[i];
  }
}

__device__ __forceinline__ float swish_f(float x) {
  return x / (1.0f + __expf(-x));
}

// ---------------- Kernel 3: fused concat + GEMM1 + swish + GEMM2 + swish ----------------
// 256 threads = 8 waves; each wave computes one 16-row tile (128 rows / block).
__global__ __launch_bounds__(256)
void mlp_kernel(const float* __restrict__ x,
                const float* __restrict__ agg,
                const float* __restrict__ u,
                const int*   __restrict__ batch,
                const float* __restrict__ W1T,
                const float* __restrict__ b1,
                const float* __restrict__ W2T,
                const float* __restrict__ b2,
                float* __restrict__ out) {
  __shared__ float hlds[8 * 16 * DIM];   // per-wave 16x64 hidden tile, 32KB

  const int lane  = threadIdx.x & 31;
  const int wave  = threadIdx.x >> 5;
  const int lrow  = lane & 15;          // N index (B/C/D) or M row (A)
  const int khalf = lane >> 4;          // selects K pair within a 4-wide K step
  const int m0    = blockIdx.x * 128 + wave * 16;

  // clamped row for this lane's A-fragment (EXEC must stay all-ones for WMMA)
  int rowA = m0 + lrow;
  if (rowA >= N_NODES) rowA = N_NODES - 1;
  const int g = batch[rowA];

  const int kh2 = khalf * 2;            // lane's K offset within a 4-wide step

  // A-source base pointers (include the lane K offset so inner loads are
  // base + immediate after unrolling)
  const float* xk = x   + (size_t)rowA * DIM + kh2;
  const float* ak = agg + (size_t)rowA * DIM + kh2;
  const float* uk = u   + (size_t)g    * DIM + kh2;

  // B base pointers: one per 16-wide N tile; W1T is [64 x 192] row-major
  const float* w1k0 = W1T + (size_t)(0 * 16 + lrow) * IN_DIM + kh2;
  const float* w1k1 = W1T + (size_t)(1 * 16 + lrow) * IN_DIM + kh2;
  const float* w1k2 = W1T + (size_t)(2 * 16 + lrow) * IN_DIM + kh2;
  const float* w1k3 = W1T + (size_t)(3 * 16 + lrow) * IN_DIM + kh2;

  v8f acc0 = {}; v8f acc1 = {}; v8f acc2 = {}; v8f acc3 = {};

  // ---- GEMM1: [16 x 192] @ [192 x 64], split by concat source ----
#define G1_STEP(SRC, KB, K0)                                                        \
  {                                                                                 \
    v2f a  = *(const v2f*)((SRC) + (K0));                                           \
    v2f b0 = *(const v2f*)(w1k0 + (KB) + (K0));                                     \
    v2f b1v = *(const v2f*)(w1k1 + (KB) + (K0));                                    \
    v2f b2v = *(const v2f*)(w1k2 + (KB) + (K0));                                    \
    v2f b3v = *(const v2f*)(w1k3 + (KB) + (K0));                                    \
    acc0 = __builtin_amdgcn_wmma_f32_16x16x4_f32(false, a, false, b0,  (short)0, acc0, false, false); \
    acc1 = __builtin_amdgcn_wmma_f32_16x16x4_f32(false, a, false, b1v, (short)0, acc1, false, false); \
    acc2 = __builtin_amdgcn_wmma_f32_16x16x4_f32(false, a, false, b2v, (short)0, acc2, false, false); \
    acc3 = __builtin_amdgcn_wmma_f32_16x16x4_f32(false, a, false, b3v, (short)0, acc3, false, false); \
  }

  #pragma unroll
  for (int k0 = 0; k0 < DIM; k0 += 4) G1_STEP(xk, 0, k0)        // K = 0..63   from x
  #pragma unroll
  for (int k0 = 0; k0 < DIM; k0 += 4) G1_STEP(ak, DIM, k0)      // K = 64..127 from agg
  #pragma unroll
  for (int k0 = 0; k0 < DIM; k0 += 4) G1_STEP(uk, 2 * DIM, k0)  // K = 128..191 from u[batch]
#undef G1_STEP

  // bias + swish, write hidden tile to LDS (row-major 16x64 per wave)
  float* hw = hlds + wave * (16 * DIM);
  {
    v8f* accs[4] = {&acc0, &acc1, &acc2, &acc3};
    #pragma unroll
    for (int t = 0; t < 4; ++t) {
      const int n = t * 16 + lrow;
      const float bias = b1[n];
      #pragma unroll
      for (int v = 0; v < 8; ++v) {
        const int m = v + 8 * khalf;
        hw[m * DIM + n] = swish_f((*accs[t])[v] + bias);
      }
    }
  }
  __syncthreads();

  // ---- GEMM2: [16 x 64] @ [64 x 64] ----
  v8f d0 = {}; v8f d1 = {}; v8f d2 = {}; v8f d3 = {};
  {
    const float* hk   = hw + lrow * DIM + kh2;                         // LDS A base
    const float* w2k0 = W2T + (size_t)(0 * 16 + lrow) * DIM + kh2;
    const float* w2k1 = W2T + (size_t)(1 * 16 + lrow) * DIM + kh2;
    const float* w2k2 = W2T + (size_t)(2 * 16 + lrow) * DIM + kh2;
    const float* w2k3 = W2T + (size_t)(3 * 16 + lrow) * DIM + kh2;
    #pragma unroll
    for (int k0 = 0; k0 < DIM; k0 += 4) {
      v2f a  = *(const v2f*)(hk + k0);
      v2f b0 = *(const v2f*)(w2k0 + k0);
      v2f b1v = *(const v2f*)(w2k1 + k0);
      v2f b2v = *(const v2f*)(w2k2 + k0);
      v2f b3v = *(const v2f*)(w2k3 + k0);
      d0 = __builtin_amdgcn_wmma_f32_16x16x4_f32(false, a, false, b0,  (short)0, d0, false, false);
      d1 = __builtin_amdgcn_wmma_f32_16x16x4_f32(false, a, false, b1v, (short)0, d1, false, false);
      d2 = __builtin_amdgcn_wmma_f32_16x16x4_f32(false, a, false, b2v, (short)0, d2, false, false);
      d3 = __builtin_amdgcn_wmma_f32_16x16x4_f32(false, a, false, b3v, (short)0, d3, false, false);
    }
  }

  // bias + swish, predicated store of output tile
  {
    v8f* accs[4] = {&d0, &d1, &d2, &d3};
    #pragma unroll
    for (int t = 0; t < 4; ++t) {
      const int n = t * 16 + lrow;
      const float bias = b2[n];
      #pragma unroll
      for (int v = 0; v < 8; ++v) {
        const int row = m0 + v + 8 * khalf;
        const float val = swish_f((*accs[t])[v] + bias);
        if (row < N_NODES) out[(size_t)row * DIM + n] = val;
      }
    }
  }
}

// ---------------- Launcher ----------------
// inputs: 0:x 1:edge_index 2:edge_attr 3:u 4:batch 5:W1 6:b1 7:W2 8:b2
extern "C" void kernel_launch(void* const* d_in, const int* in_sizes, int n_in,
                              void* d_out, int out_size, void* d_ws, size_t ws_size,
                              hipStream_t stream) {
  const float* x         = (const float*)d_in[0];
  const int*   edge_idx  = (const int*)d_in[1];     // [2, N_EDGES]; col = row 1
  const float* edge_attr = (const float*)d_in[2];
  const float* u         = (const float*)d_in[3];
  const int*   batch     = (const int*)d_in[4];
  const float* W1        = (const float*)d_in[5];
  const float* b1        = (const float*)d_in[6];
  const float* W2        = (const float*)d_in[7];
  const float* b2        = (const float*)d_in[8];
  float* out = (float*)d_out;

  // workspace layout (floats): agg [N_NODES*DIM], W1T [DIM*IN_DIM], W2T [DIM*DIM]
  float* agg = (float*)d_ws;
  float* W1T = agg + (size_t)N_NODES * DIM;
  float* W2T = W1T + (size_t)DIM * IN_DIM;

  const int aggN = N_NODES * DIM;           // 3,200,000 floats
  const int n4   = aggN / 4;
  zero_kernel<<<(n4 + 255) / 256, 256, 0, stream>>>((float4*)agg, n4);

  const int scatterN = N_EDGES * 16;        // (edge, float4-chunk)
  scatter_kernel<<<(scatterN + 255) / 256, 256, 0, stream>>>(
      edge_attr, edge_idx + N_EDGES /*col*/, agg);

  transpose_kernel<<<(IN_DIM * DIM + 255) / 256, 256, 0, stream>>>(W1, W2, W1T, W2T);

  const int blocks = (N_NODES + 127) / 128; // 128 rows per block (8 waves x 16)
  mlp_kernel<<<blocks, 256, 0, stream>>>(x, agg, u, batch, W1T, b1, W2T, b2, out);
}